// FeatureDecoder_20255065768568
// MI455X (gfx1250) — compile-verified
//
#include <hip/hip_runtime.h>

// GCN 3-layer forward for MI455X (gfx1250, wave32).
//
// - fp32 WMMA (v_wmma_f32_16x16x4_f32) for the GEMMs; W column-strip staged in
//   LDS (stride 72 dwords -> the two 16-lane wave halves hit disjoint bank
//   sets: 2*72 mod 64 == 16). Each wave computes a 16x32 strip (2 C tiles)
//   sharing one A fragment.
// - Edge scatter is the dominant cost; h (51MB) and out (51MB) are L2-resident
//   (192MB L2). Coalesced float4 gathers + coalesced f32 atomics; edge-index
//   streams are loaded non-temporally so they don't evict the feature tiles.
// - Bias folded into output init; ReLU folded into next layer's A-load.
// - Degrees via integer atomics (deterministic).

typedef __attribute__((ext_vector_type(2))) float v2f;
typedef __attribute__((ext_vector_type(8))) float v8f;

// ---------------- degree / norm ----------------

__global__ __launch_bounds__(256) void zero_u32_kernel(unsigned* __restrict__ p, int n) {
  int i = blockIdx.x * blockDim.x + threadIdx.x;
  if (i < n) p[i] = 0u;
}

__global__ __launch_bounds__(256) void deg_count_kernel(const int* __restrict__ dst,
                                                        unsigned* __restrict__ cnt, int E) {
  int i = blockIdx.x * blockDim.x + threadIdx.x;
  if (i < E) {
    int d = __builtin_nontemporal_load(dst + i);  // streamed once: don't pollute L2
    atomicAdd(&cnt[d], 1u);
  }
}

__global__ __launch_bounds__(256) void dinv_kernel(const unsigned* __restrict__ cnt,
                                                   float* __restrict__ dinv, int n) {
  int i = blockIdx.x * blockDim.x + threadIdx.x;
  if (i < n) dinv[i] = rsqrtf((float)cnt[i] + 1.0f);  // +1 = self loop; always > 0
}

// ---------------- bias broadcast init (out = b, scatter adds on top) ----------------

__global__ __launch_bounds__(256) void bias_init_kernel(float* __restrict__ out,
                                                        const float* __restrict__ b,
                                                        int total, int cmask) {
  int i = blockIdx.x * blockDim.x + threadIdx.x;
  if (i < total) out[i] = b[i & cmask];  // C is a power of two (256/128/64)
}

// ---------------- fp32 WMMA GEMM: H[M,N] = act(X[M,K]) @ W[K,N] ----------------
// Block = 8 wave32s as 4(M) x 2(N) wave tiles of 16x32 => 64x64 block tile.
// W[:, nblk:nblk+64] staged in LDS once per block (row stride 72 dwords).
// A frag (16x4 f32, 2 VGPR): lane L holds A[L&15][k0 + 2*(L>>4) + j], j=reg.
// B frag (4x16, 2 VGPR):     reg j, lane L -> B row (k0 + 2*(L>>4) + j), col L&15.
// C/D (16x16, 8 VGPR):       reg r, lane L -> C[r + 8*(L>>4)][n0 + (L&15)].

#define BLDS_STRIDE 72  // dwords; 2*72 mod 64 = 16 -> conflict-free wave halves

template <bool RELU>
__global__ __launch_bounds__(256) void gemm_wmma_kernel(const float* __restrict__ X,
                                                        const float* __restrict__ W,
                                                        float* __restrict__ H,
                                                        int M, int K, int N) {
  extern __shared__ float Blds[];  // [K][BLDS_STRIDE], first 64 cols used
  const int tid  = threadIdx.y * 32 + threadIdx.x;
  const int nblk = blockIdx.y * 64;

  // Cooperative, coalesced fill of the W column strip (all waves participate).
  for (int idx = tid; idx < K * 64; idx += 256) {
    const int k = idx >> 6, c = idx & 63;
    Blds[k * BLDS_STRIDE + c] = W[(size_t)k * (size_t)N + (nblk + c)];
  }
  __syncthreads();

  const int lane  = threadIdx.x;   // 0..31
  const int wave  = threadIdx.y;   // 0..7
  const int wm    = wave >> 1;     // 0..3
  const int wn    = wave & 1;      // 0..1
  const int m0    = blockIdx.x * 64 + wm * 16;
  const int l15   = lane & 15;
  const int khalf = lane >> 4;     // 0 or 1

  if (m0 < M) {  // wave-uniform guard: EXEC all-ones at every WMMA
    const size_t arow = (size_t)(m0 + l15) * (size_t)K;
    const int    nc0  = wn * 32 + l15;  // LDS column of tile 0
    v8f acc0 = {}, acc1 = {};

    for (int k0 = 0; k0 < K; k0 += 4) {
      const int ka = k0 + 2 * khalf;
      float a0 = X[arow + ka];
      float a1 = X[arow + ka + 1];
      if (RELU) { a0 = fmaxf(a0, 0.0f); a1 = fmaxf(a1, 0.0f); }
      v2f a; a.x = a0; a.y = a1;

      const float* bp = &Blds[ka * BLDS_STRIDE + nc0];
      v2f b0; b0.x = bp[0];  b0.y = bp[BLDS_STRIDE];
      v2f b1; b1.x = bp[16]; b1.y = bp[BLDS_STRIDE + 16];

      acc0 = __builtin_amdgcn_wmma_f32_16x16x4_f32(false, a, false, b0,
                                                   (short)0, acc0, false, false);
      acc1 = __builtin_amdgcn_wmma_f32_16x16x4_f32(false, a, false, b1,
                                                   (short)0, acc1, false, false);
    }

    const int colbase = nblk + wn * 32 + l15;
#pragma unroll
    for (int r = 0; r < 8; ++r) {
      const size_t row = (size_t)(m0 + r + 8 * khalf) * (size_t)N;
      H[row + colbase]      = acc0[r];
      H[row + colbase + 16] = acc1[r];
    }
  }
}

// ---------------- edge scatter: out[dst] += h[src] * dinv[src]*dinv[dst] ----------------
// One thread handles 4 consecutive features (b128 gather) of one edge; the
// threads of one edge are consecutive -> coalesced gathers and atomics.
// Edges e in [0,E) come from edge_index; e in [E, E+Nn) are self loops.

__global__ __launch_bounds__(256) void scatter_kernel(const float* __restrict__ H,
                                                      const int* __restrict__ src,
                                                      const int* __restrict__ dst,
                                                      const float* __restrict__ dinv,
                                                      float* __restrict__ out,
                                                      int E, int Nn, int C, int logChunks) {
  const int idx = blockIdx.x * blockDim.x + threadIdx.x;
  const int total = (E + Nn) << logChunks;
  if (idx >= total) return;
  const int e = idx >> logChunks;
  const int c = (idx & ((1 << logChunks) - 1)) << 2;  // feature offset (x4 floats)

  int s, d;
  float nrm;
  if (e < E) {
    s = __builtin_nontemporal_load(src + e);  // index streams: keep L2 for features
    d = __builtin_nontemporal_load(dst + e);
    nrm = dinv[s] * dinv[d];
  } else {
    s = d = e - E;
    const float di = dinv[s];
    nrm = di * di;
  }

  const float4 v = *reinterpret_cast<const float4*>(H + (size_t)s * (size_t)C + c);
  float* o = out + (size_t)d * (size_t)C + c;
  atomicAdd(o + 0, v.x * nrm);
  atomicAdd(o + 1, v.y * nrm);
  atomicAdd(o + 2, v.z * nrm);
  atomicAdd(o + 3, v.w * nrm);
}

// ---------------- launcher ----------------

extern "C" void kernel_launch(void* const* d_in, const int* in_sizes, int n_in,
                              void* d_out, int out_size, void* d_ws, size_t ws_size,
                              hipStream_t stream) {
  const float* z  = (const float*)d_in[0];
  const int*   ei = (const int*)d_in[1];     // [2, E] row-major: src then dst
  const float* W0 = (const float*)d_in[2];
  const float* b0 = (const float*)d_in[3];
  const float* W1 = (const float*)d_in[4];
  const float* b1 = (const float*)d_in[5];
  const float* W2 = (const float*)d_in[6];
  const float* b2 = (const float*)d_in[7];

  const int D0 = 128, D1 = 256, D2 = 128, D3 = 64;
  const int Nn = in_sizes[0] / D0;           // 50000
  const int E  = in_sizes[1] / 2;            // 1600000
  const int* srcp = ei;
  const int* dstp = ei + E;

  // Workspace layout (floats): [cnt(u32) | dinv | hbuf(Nn*256) | obuf(Nn*256)] ~103MB
  const size_t nPad = ((size_t)Nn + 63) & ~(size_t)63;
  unsigned* cnt  = (unsigned*)d_ws;
  float*    dinv = (float*)d_ws + nPad;
  float*    hbuf = (float*)d_ws + 2 * nPad;
  float*    obuf = hbuf + (size_t)Nn * 256;

  const int TB = 256;

  // degrees -> dinv (deterministic integer atomics)
  zero_u32_kernel<<<(Nn + TB - 1) / TB, TB, 0, stream>>>(cnt, Nn);
  deg_count_kernel<<<(E + TB - 1) / TB, TB, 0, stream>>>(dstp, cnt, E);
  dinv_kernel<<<(Nn + TB - 1) / TB, TB, 0, stream>>>(cnt, dinv, Nn);

  auto run_layer = [&](const float* x, const float* Wm, const float* bias,
                       float* h, float* out, int K, int C, bool relu_in) {
    dim3 blk(32, 8);
    dim3 grd((unsigned)((Nn + 63) / 64), (unsigned)(C / 64));
    const size_t shbytes = (size_t)K * BLDS_STRIDE * sizeof(float);  // <= 73728B
    if (relu_in)
      gemm_wmma_kernel<true><<<grd, blk, shbytes, stream>>>(x, Wm, h, Nn, K, C);
    else
      gemm_wmma_kernel<false><<<grd, blk, shbytes, stream>>>(x, Wm, h, Nn, K, C);

    const int total = Nn * C;
    bias_init_kernel<<<(total + TB - 1) / TB, TB, 0, stream>>>(out, bias, total, C - 1);

    const int logc = (C == 256) ? 6 : (C == 128) ? 5 : 4;  // log2(C/4)
    const int sThreads = (E + Nn) << logc;
    scatter_kernel<<<(sThreads + TB - 1) / TB, TB, 0, stream>>>(
        h, srcp, dstp, dinv, out, E, Nn, C, logc);
  };

  // layer 0: z(128) -> 256 ; layer 1: relu(out0)(256) -> 128 ; layer 2: relu(out1)(128) -> 64
  run_layer(z,    W0, b0, hbuf, obuf,           D0, D1, false);
  run_layer(obuf, W1, b1, hbuf, obuf,           D1, D2, true);
  run_layer(obuf, W2, b2, hbuf, (float*)d_out,  D2, D3, true);
}